// MultiObjectContrastiveAttnLoss_13683765805564
// MI455X (gfx1250) — compile-verified
//
#include <hip/hip_runtime.h>

typedef __attribute__((ext_vector_type(2))) float v2f;
typedef __attribute__((ext_vector_type(8))) float v8f;

#define GAMMA1 5.0f
#define GAMMA2 5.0f
#define GAMMA3 10.0f
#define MIN_VAL 1e-6f

constexpr int BS = 128;   // batch
constexpr int TL = 32;    // "txt" length (img_embs spans)
constexpr int IL = 64;    // "img" length (txt_embs tokens)
constexpr int D  = 256;   // embed dim
constexpr int DP = D + 4; // padded LDS row stride (floats) for 256-wide rows
constexpr int SL = IL + 4;// padded LDS row stride for sim/alphas rows

// ---------------------------------------------------------------------------
// Kernel 1: row-wise L2 normalization (one wave32 per 256-float row)
// ---------------------------------------------------------------------------
__global__ void norm_rows_kernel(const float* __restrict__ a,   // img_embs (plays txt)
                                 const float* __restrict__ b,   // txt_embs (plays img)
                                 float* __restrict__ ntxt,
                                 float* __restrict__ nimg) {
  int wave = threadIdx.x >> 5;
  int lane = threadIdx.x & 31;
  int row  = blockIdx.x * 8 + wave;           // 12288 rows total
  const float* src;
  float* dst;
  if (row < BS * TL) {
    src = a + (size_t)row * D;
    dst = ntxt + (size_t)row * D;
  } else {
    int r = row - BS * TL;
    src = b + (size_t)r * D;
    dst = nimg + (size_t)r * D;
  }
  float vals[8];
  float ss = 0.f;
#pragma unroll
  for (int k = 0; k < 8; ++k) {
    vals[k] = src[lane + k * 32];
    ss += vals[k] * vals[k];
  }
#pragma unroll
  for (int off = 16; off; off >>= 1) ss += __shfl_xor(ss, off, 32);
  float inv = 1.0f / fmaxf(sqrtf(ss), 1e-12f);
#pragma unroll
  for (int k = 0; k < 8; ++k) dst[lane + k * 32] = vals[k] * inv;
}

// ---------------------------------------------------------------------------
// Kernel 2: fused per-pair (i,j) kernel. 256 threads = 8 wave32.
//   sim  = ntxt_i(32x256) x nimg_j^T   via V_WMMA_F32_16X16X4_F32
//   span-normalize -> alphas -> contexts = alphas x nimg_j (WMMA again)
//   l2norm(contexts), diag dot, log-sum-exp -> sample[i,j]
//   i==j: emit alphas into out (transposed to (il,tl))
// ---------------------------------------------------------------------------
__global__ __launch_bounds__(256, 1)
void pair_kernel(const float* __restrict__ ntxt, const float* __restrict__ nimg,
                 const int* __restrict__ tlen_arr, const int* __restrict__ ilen_arr,
                 float* __restrict__ sample, float* __restrict__ out_alphas) {
  extern __shared__ float smem[];
  float* s_txt = smem;                 // TL x DP
  float* s_img = s_txt + TL * DP;      // IL x DP
  float* s_sim = s_img + IL * DP;      // TL x SL   (sim, then alphas)
  float* s_ctx = s_sim + TL * SL;      // TL x DP
  float* s_red = s_ctx + TL * DP;      // 64 scratch

  const int j = blockIdx.x;
  const int i = blockIdx.y;
  const int tid = threadIdx.x;
  const int wave = tid >> 5;
  const int lane = tid & 31;
  const int hm   = lane & 15;   // m (or n) within 16
  const int half = lane >> 4;   // k-pair selector

  const int tlen = tlen_arr[i];
  const int ilen = ilen_arr[j];

  // ---- stage tiles into LDS (padded rows), float4 wide ----
  {
    const float4* gt = (const float4*)(ntxt + (size_t)i * TL * D);
    for (int idx = tid; idx < TL * D / 4; idx += 256) {
      int r = idx / (D / 4), c = idx % (D / 4);
      ((float4*)(s_txt + r * DP))[c] = gt[idx];
    }
    const float4* gi = (const float4*)(nimg + (size_t)j * IL * D);
    for (int idx = tid; idx < IL * D / 4; idx += 256) {
      int r = idx / (D / 4), c = idx % (D / 4);
      ((float4*)(s_img + r * DP))[c] = gi[idx];
    }
  }
  __syncthreads();

  // ---- phase A: sim(32x64) = ntxt x nimg^T ; one 16x16 tile per wave ----
  // Two interleaved accumulator chains break the WMMA RAW hazard chain.
  {
    const int mt = wave >> 2, nt = wave & 3;
    const int M0 = mt * 16, N0 = nt * 16;
    v8f acc0 = {}, acc1 = {};
    const float* arow = s_txt + (M0 + hm) * DP;  // A: row m, K pair at 2*half
    const float* brow = s_img + (N0 + hm) * DP;  // B[k][n] = nimg[n][k]
    for (int k0 = 0; k0 < D; k0 += 8) {
      int ka = k0 + 2 * half;
      int kb = k0 + 4 + 2 * half;
      v2f a0 = {arow[ka], arow[ka + 1]};
      v2f b0 = {brow[ka], brow[ka + 1]};
      v2f a1 = {arow[kb], arow[kb + 1]};
      v2f b1 = {brow[kb], brow[kb + 1]};
      acc0 = __builtin_amdgcn_wmma_f32_16x16x4_f32(false, a0, false, b0,
                                                   (short)0, acc0, false, false);
      acc1 = __builtin_amdgcn_wmma_f32_16x16x4_f32(false, a1, false, b1,
                                                   (short)0, acc1, false, false);
    }
#pragma unroll
    for (int r = 0; r < 8; ++r)
      s_sim[(M0 + r + 8 * half) * SL + (N0 + hm)] = acc0[r] + acc1[r];
  }
  __syncthreads();

  // ---- phase B: masked exp, span-normalize over t, alphas over l ----
  for (int idx = tid; idx < TL * IL; idx += 256) {
    int t = idx >> 6, l = idx & 63;
    bool m = (t >= tlen) || (l >= ilen);
    float v = s_sim[t * SL + l];
    s_sim[t * SL + l] = m ? 0.f : expf(v);
  }
  __syncthreads();
  if (tid < IL) {            // denom over spans t, per column l
    float s = 0.f;
    for (int t = 0; t < TL; ++t) s += s_sim[t * SL + tid];
    s_red[tid] = s + MIN_VAL;
  }
  __syncthreads();
  for (int idx = tid; idx < TL * IL; idx += 256) {
    int t = idx >> 6, l = idx & 63;
    bool m = (t >= tlen) || (l >= ilen);
    float v = s_sim[t * SL + l] / s_red[l];
    s_sim[t * SL + l] = m ? 0.f : expf(GAMMA1 * v);
  }
  __syncthreads();
  if (tid < TL) {            // alpha denom over image tokens l, per row t
    float s = 0.f;
    for (int l = 0; l < IL; ++l) s += s_sim[tid * SL + l];
    s_red[tid] = s + MIN_VAL;
  }
  __syncthreads();
  for (int idx = tid; idx < TL * IL; idx += 256) {
    int t = idx >> 6, l = idx & 63;
    s_sim[t * SL + l] = s_sim[t * SL + l] / s_red[t];
  }
  __syncthreads();

  if (i == j) {              // alphas_diag[b, l, t] = alphas[b, b, t, l]
    for (int idx = tid; idx < TL * IL; idx += 256) {
      int t = idx >> 6, l = idx & 63;
      out_alphas[(size_t)i * (TL * IL) + l * TL + t] = s_sim[t * SL + l];
    }
  }

  // ---- phase C: contexts(32x256) = alphas(32x64) x nimg(64x256) ----
  // Each wave owns N-tile pair {wave, wave+8} for both M-tiles: the A-frag is
  // shared between the two B-frags, and the two accumulators form independent
  // WMMA chains.
#pragma unroll
  for (int p = 0; p < 2; ++p) {          // p = M-tile
    const int M0 = p * 16;
    const int n0 = wave * 16;            // N-tile A
    const int n1 = (wave + 8) * 16;      // N-tile B
    v8f acc0 = {}, acc1 = {};
    const float* arow = s_sim + (M0 + hm) * SL;
    for (int k0 = 0; k0 < IL; k0 += 4) {
      int k = k0 + 2 * half;
      v2f a  = {arow[k], arow[k + 1]};
      v2f b0 = {s_img[k * DP + n0 + hm], s_img[(k + 1) * DP + n0 + hm]};
      v2f b1 = {s_img[k * DP + n1 + hm], s_img[(k + 1) * DP + n1 + hm]};
      acc0 = __builtin_amdgcn_wmma_f32_16x16x4_f32(false, a, false, b0,
                                                   (short)0, acc0, false, false);
      acc1 = __builtin_amdgcn_wmma_f32_16x16x4_f32(false, a, false, b1,
                                                   (short)0, acc1, false, false);
    }
#pragma unroll
    for (int r = 0; r < 8; ++r) {
      s_ctx[(M0 + r + 8 * half) * DP + (n0 + hm)] = acc0[r];
      s_ctx[(M0 + r + 8 * half) * DP + (n1 + hm)] = acc1[r];
    }
  }
  __syncthreads();

  // ---- phase D: l2norm(ctx) rows, diag dot, masked exp; 8 lanes per row ----
  {
    int t = tid >> 3;         // 0..31
    int seg = tid & 7;        // 0..7  (32 elems each)
    const float* crow = s_ctx + t * DP;
    const float* trow = s_txt + t * DP;
    float ss = 0.f, dp = 0.f;
    for (int k = seg * 32; k < seg * 32 + 32; ++k) {
      float c = crow[k];
      ss += c * c;
      dp += c * trow[k];
    }
#pragma unroll
    for (int off = 4; off; off >>= 1) {
      ss += __shfl_down(ss, off, 8);
      dp += __shfl_down(dp, off, 8);
    }
    if (seg == 0) {
      float inv  = 1.0f / fmaxf(sqrtf(ss), 1e-12f);
      float diag = dp * inv;
      s_red[t] = (t < tlen) ? expf(GAMMA2 * diag) : 0.f;
    }
  }
  __syncthreads();
  if (tid == 0) {
    float s = 0.f;
    for (int t = 0; t < TL; ++t) s += s_red[t];
    sample[(size_t)i * BS + j] = logf(s) / GAMMA2;   // (1/G2)*log(sum exp(G2*diag))
  }
}

// ---------------------------------------------------------------------------
// Kernel 3: contrastive loss over the 128x128 sample matrix -> out[0]
// ---------------------------------------------------------------------------
__global__ void loss_kernel(const float* __restrict__ sample, float* __restrict__ out) {
  __shared__ float s_loss[BS];
  int tid = threadIdx.x;  // 128 threads
  float rowsum = 0.f, colsum = 0.f;
  for (int k = 0; k < BS; ++k) {
    rowsum += expf(GAMMA3 * sample[tid * BS + k]);   // sum over j (axis -1)
    colsum += expf(GAMMA3 * sample[k * BS + tid]);   // sum over i (axis 0)
  }
  float p = expf(GAMMA3 * sample[tid * BS + tid]);
  float loss = -logf(p / (colsum + MIN_VAL)) - logf(p / (rowsum + MIN_VAL));
  s_loss[tid] = loss;
  __syncthreads();
  if (tid == 0) {
    float s = 0.f;
    for (int k = 0; k < BS; ++k) s += s_loss[k];
    out[0] = s / (float)BS;
  }
}

// ---------------------------------------------------------------------------
extern "C" void kernel_launch(void* const* d_in, const int* in_sizes, int n_in,
                              void* d_out, int out_size, void* d_ws, size_t ws_size,
                              hipStream_t stream) {
  const float* img_embs = (const float*)d_in[0];  // plays "txt" in _word_loss
  const float* txt_embs = (const float*)d_in[1];  // plays "img"
  const int*   img_lens = (const int*)d_in[2];    // plays txt_len
  const int*   txt_lens = (const int*)d_in[3];    // plays img_len

  float* out = (float*)d_out;     // [0] = mean loss, [1..] = alphas_diag (128x64x32)
  float* ws  = (float*)d_ws;
  float* ntxt   = ws;                         // 128*32*256
  float* nimg   = ntxt + (size_t)BS * TL * D; // 128*64*256
  float* sample = nimg + (size_t)BS * IL * D; // 128*128

  norm_rows_kernel<<<(BS * TL + BS * IL) / 8, 256, 0, stream>>>(
      img_embs, txt_embs, ntxt, nimg);

  size_t lds_bytes = (size_t)(TL * DP + IL * DP + TL * SL + TL * DP + 64) * sizeof(float);
  dim3 grid(BS, BS);  // x = j, y = i
  pair_kernel<<<grid, 256, lds_bytes, stream>>>(
      ntxt, nimg, img_lens, txt_lens, sample, out + 1);

  loss_kernel<<<1, BS, 0, stream>>>(sample, out);
}